// SmoothLDDTLoss_33921651704340
// MI455X (gfx1250) — compile-verified
//
#include <hip/hip_runtime.h>
#include <math.h>

typedef __attribute__((ext_vector_type(2))) float v2f;
typedef __attribute__((ext_vector_type(8))) float v8f;

#define N_ATOMS 8192
#define NT      512                     // 16x16 tiles per dimension
#define NTRI    (NT * (NT + 1) / 2)     // 131328 upper-tri tiles
#define TILES_PER_WAVE 8
#define WAVES_PER_BLOCK 8
#define NBLOCKS (NTRI / (TILES_PER_WAVE * WAVES_PER_BLOCK))   // 2052 exactly
#define RADIUS_F 15.0f

// start tile index of triangular row r: r*NT - r*(r-1)/2
__device__ __forceinline__ int tri_base(int r) { return (r * (2 * NT + 1 - r)) >> 1; }

__global__ void lddt_zero_acc(float* acc) {
    if (threadIdx.x < 2) acc[threadIdx.x] = 0.0f;
}

__global__ __launch_bounds__(256) void lddt_tiles_kernel(const float* __restrict__ predC,
                                                         const float* __restrict__ trueC,
                                                         float* __restrict__ acc) {
    const int lane   = threadIdx.x & 31;
    const int waveId = threadIdx.x >> 5;
    const int tStart = (blockIdx.x * WAVES_PER_BLOCK + waveId) * TILES_PER_WAVE;

    const int l16 = lane & 15;
    const int kh  = lane >> 4;   // K-half: 0 -> K=0,1 ; 1 -> K=2,3

    // constants: exp(-2t) for t in {0.5, 1, 2, 4}
    const float c1 = 0.36787944117144233f;     // e^-1
    const float c2 = 0.13533528323661270f;     // e^-2
    const float c4 = 0.01831563888873418f;     // e^-4
    const float c8 = 3.3546262790251185e-4f;   // e^-8

    float wScore = 0.0f, wMask = 0.0f;   // per-wave (per-lane partial) totals

    for (int t = tStart; t < tStart + TILES_PER_WAVE; ++t) {
        // ---- decode triangular tile index -> (r, c), r <= c ----
        int r = (int)((1025.0f - sqrtf((float)(1025 * 1025 - 8 * t))) * 0.5f);
        if (r < 0) r = 0;
        if (r > NT - 1) r = NT - 1;
        while (tri_base(r + 1) <= t) ++r;
        while (tri_base(r) > t) --r;
        const int   c = r + (t - tri_base(r));
        const float w = (r == c) ? 1.0f : 2.0f;   // symmetry weight

        // ---- per-lane coordinate loads (row for A, col for B) ----
        const int rowIdx = (r << 4) + l16;
        const int colIdx = (c << 4) + l16;

        const float prx = predC[rowIdx * 3 + 0], pry = predC[rowIdx * 3 + 1], prz = predC[rowIdx * 3 + 2];
        const float pcx = predC[colIdx * 3 + 0], pcy = predC[colIdx * 3 + 1], pcz = predC[colIdx * 3 + 2];
        const float qrx = trueC[rowIdx * 3 + 0], qry = trueC[rowIdx * 3 + 1], qrz = trueC[rowIdx * 3 + 2];
        const float qcx = trueC[colIdx * 3 + 0], qcy = trueC[colIdx * 3 + 1], qcz = trueC[colIdx * 3 + 2];

        const float sqProw = prx * prx + pry * pry + prz * prz;
        const float sqPcol = pcx * pcx + pcy * pcy + pcz * pcz;
        const float sqTrow = qrx * qrx + qry * qry + qrz * qrz;
        const float sqTcol = qcx * qcx + qcy * qcy + qcz * qcz;

        // ---- WMMA computes d^2 directly:  D = A x B + C
        //   A[m][:] = ( x_m,  y_m,  z_m, sq_m )      (K=3 slot carries row norm)
        //   B[:][n] = (-2x_n, -2y_n, -2z_n, 1 )
        //   C[m][n] = sq_n  (lane-constant broadcast: each lane's col norm)
        // => D[m][n] = sq_m + sq_n - 2 * x_m . x_n
        v2f aP, bP, aT, bT;
        aP.x = kh ? prz : prx;             aP.y = kh ? sqProw : pry;
        bP.x = kh ? (-2.0f * pcz) : (-2.0f * pcx);
        bP.y = kh ? 1.0f          : (-2.0f * pcy);
        aT.x = kh ? qrz : qrx;             aT.y = kh ? sqTrow : qry;
        bT.x = kh ? (-2.0f * qcz) : (-2.0f * qcx);
        bT.y = kh ? 1.0f          : (-2.0f * qcy);

        v8f cP, cT;
#pragma unroll
        for (int i = 0; i < 8; ++i) { cP[i] = sqPcol; cT[i] = sqTcol; }

        v8f d2P = __builtin_amdgcn_wmma_f32_16x16x4_f32(false, aP, false, bP,
                                                        (short)0, cP, false, false);
        v8f d2T = __builtin_amdgcn_wmma_f32_16x16x4_f32(false, aT, false, bT,
                                                        (short)0, cT, false, false);

        float sScore = 0.0f, sMask = 0.0f;
#pragma unroll
        for (int v = 0; v < 8; ++v) {
            const float dp = __builtin_amdgcn_sqrtf(fmaxf(d2P[v], 0.0f));
            const float dt = __builtin_amdgcn_sqrtf(fmaxf(d2T[v], 0.0f));

            const bool  m    = (dt < RADIUS_F) && (dt > 0.0f);
            const float diff = fabsf(dp - dt);
            // sigmoid(2(t-diff)) = 1 / (1 + e^{2*diff} * e^{-2t}) -> single exp
            const float e = __expf(2.0f * diff);
            const float s = 0.25f * (__builtin_amdgcn_rcpf(fmaf(e, c1, 1.0f)) +
                                     __builtin_amdgcn_rcpf(fmaf(e, c2, 1.0f)) +
                                     __builtin_amdgcn_rcpf(fmaf(e, c4, 1.0f)) +
                                     __builtin_amdgcn_rcpf(fmaf(e, c8, 1.0f)));
            sScore += m ? s : 0.0f;
            sMask  += m ? 1.0f : 0.0f;
        }
        wScore += w * sScore;
        wMask  += w * sMask;
    }

    // ---- wave32 reduction, then 2 atomics per wave (amortized over 8 tiles) ----
#pragma unroll
    for (int off = 16; off > 0; off >>= 1) {
        wScore += __shfl_xor(wScore, off, 32);
        wMask  += __shfl_xor(wMask,  off, 32);
    }
    if (lane == 0) {
        atomicAdd(&acc[0], wScore);
        atomicAdd(&acc[1], wMask);
    }
}

__global__ void lddt_finalize(const float* __restrict__ acc, float* __restrict__ out) {
    const float num   = acc[0];
    const float denom = acc[1];
    const float lddt  = (denom > 0.0f) ? (num / fmaxf(denom, 1.0f)) : 0.5f;
    out[0] = 1.0f - lddt;
    out[1] = lddt;
}

extern "C" void kernel_launch(void* const* d_in, const int* in_sizes, int n_in,
                              void* d_out, int out_size, void* d_ws, size_t ws_size,
                              hipStream_t stream) {
    (void)in_sizes; (void)n_in; (void)out_size; (void)ws_size;
    const float* predC = (const float*)d_in[0];
    const float* trueC = (const float*)d_in[1];
    float*       out   = (float*)d_out;
    float*       acc   = (float*)d_ws;

    lddt_zero_acc<<<1, 32, 0, stream>>>(acc);
    // 131328 tiles = 2052 blocks x 8 waves x 8 tiles/wave
    lddt_tiles_kernel<<<NBLOCKS, 256, 0, stream>>>(predC, trueC, acc);
    lddt_finalize<<<1, 1, 0, stream>>>(acc, out);
}